// PointerGeneratorDecoder_18210661335340
// MI455X (gfx1250) — compile-verified
//
#include <hip/hip_runtime.h>
#include <math.h>

// ---- problem constants (match reference) ----
#define V_SZ   50000
#define H_SZ   1024
#define L_SZ   400
#define B_SZ   64
#define OOV    500
#define ROW_P  (V_SZ + OOV)          // 50500, p_final row stride

// d_out layout (floats): p_final [B,V+500] | hidden [B,H] | attn_weights [B,L]
#define OUT_P  0
#define OUT_H  (B_SZ * ROW_P)                    // 3,232,000
#define OUT_A  (OUT_H + B_SZ * H_SZ)             // 3,297,536

// workspace layout (floats)
#define WS_EMB   0
#define WS_ATTN  (WS_EMB  + B_SZ * H_SZ)
#define WS_APP   (WS_ATTN + B_SZ * L_SZ)
#define WS_COMB  (WS_APP  + B_SZ * H_SZ)
#define WS_HNEW  (WS_COMB + B_SZ * H_SZ)
#define WS_PGEN  (WS_HNEW + B_SZ * H_SZ)
#define WS_LSE   (WS_PGEN + B_SZ)

typedef __attribute__((ext_vector_type(2))) float v2f;
typedef __attribute__((ext_vector_type(8))) float v8f;

__device__ __forceinline__ float sigmoidf_(float x) { return 1.0f / (1.0f + __expf(-x)); }

// ---------------- K1: embed gather + attn logits + softmax ----------------
__global__ __launch_bounds__(256) void k1_front_attn(
    const int* __restrict__ input_ids, const float* __restrict__ hidden,
    const float* __restrict__ emb, const float* __restrict__ attn_W,
    const float* __restrict__ attn_b, float* __restrict__ ws, float* __restrict__ out) {
  const int b = blockIdx.x, tid = threadIdx.x;
  __shared__ float s_cat[2 * H_SZ];
  __shared__ float s_logit[L_SZ];
  __shared__ float red[256];

  const int tok = input_ids[b];
  for (int j = tid; j < H_SZ; j += 256) {
    float e = emb[(size_t)tok * H_SZ + j];
    s_cat[j] = e;
    ws[WS_EMB + b * H_SZ + j] = e;
    s_cat[H_SZ + j] = hidden[b * H_SZ + j];
  }
  __syncthreads();

  for (int l = tid; l < L_SZ; l += 256) {
    const float* w = attn_W + (size_t)l * 2 * H_SZ;
    float acc = attn_b[l];
    #pragma unroll 4
    for (int j = 0; j < 2 * H_SZ; ++j) acc = fmaf(s_cat[j], w[j], acc);
    s_logit[l] = acc;
  }
  __syncthreads();

  // softmax over L
  float m = -INFINITY;
  for (int l = tid; l < L_SZ; l += 256) m = fmaxf(m, s_logit[l]);
  red[tid] = m; __syncthreads();
  for (int s = 128; s > 0; s >>= 1) { if (tid < s) red[tid] = fmaxf(red[tid], red[tid + s]); __syncthreads(); }
  const float mx = red[0]; __syncthreads();

  float sm = 0.f;
  for (int l = tid; l < L_SZ; l += 256) { float e = __expf(s_logit[l] - mx); s_logit[l] = e; sm += e; }
  red[tid] = sm; __syncthreads();
  for (int s = 128; s > 0; s >>= 1) { if (tid < s) red[tid] += red[tid + s]; __syncthreads(); }
  const float inv = 1.0f / red[0];

  for (int l = tid; l < L_SZ; l += 256) {
    float w = s_logit[l] * inv;
    ws[WS_ATTN + b * L_SZ + l] = w;
    out[OUT_A + b * L_SZ + l]  = w;   // third tuple output
  }
}

// ---------------- K2: attn_applied[b,h] = sum_l w[b,l] * enc[l,b,h] -------
__global__ __launch_bounds__(256) void k2_attn_apply(
    const float* __restrict__ enc, const float* __restrict__ ws_in, float* __restrict__ ws) {
  const int b = blockIdx.y;
  const int h = blockIdx.x * 256 + threadIdx.x;
  const float* aw = ws_in + WS_ATTN + b * L_SZ;
  float acc = 0.f;
  #pragma unroll 4
  for (int l = 0; l < L_SZ; ++l)
    acc = fmaf(aw[l], enc[((size_t)l * B_SZ + b) * H_SZ + h], acc);
  ws[WS_APP + b * H_SZ + h] = acc;
}

// ---------------- K3: combined = [emb|applied] @ comb_W^T + comb_b -------
__global__ __launch_bounds__(256) void k3_combine(
    const float* __restrict__ comb_W, const float* __restrict__ comb_b,
    float* __restrict__ ws) {
  const int b = blockIdx.y;
  const int h = blockIdx.x * 256 + threadIdx.x;
  __shared__ float s_cat[2 * H_SZ];
  for (int j = threadIdx.x; j < H_SZ; j += 256) {
    s_cat[j]        = ws[WS_EMB + b * H_SZ + j];
    s_cat[H_SZ + j] = ws[WS_APP + b * H_SZ + j];
  }
  __syncthreads();
  const float* w = comb_W + (size_t)h * 2 * H_SZ;
  float acc = comb_b[h];
  #pragma unroll 4
  for (int j = 0; j < 2 * H_SZ; ++j) acc = fmaf(s_cat[j], w[j], acc);
  ws[WS_COMB + b * H_SZ + h] = acc;
}

// ---------------- K4: single-step GRU -> h_new ----------------------------
__global__ __launch_bounds__(256) void k4_gru(
    const float* __restrict__ W_ih, const float* __restrict__ W_hh,
    const float* __restrict__ b_ih, const float* __restrict__ b_hh,
    const float* __restrict__ hidden, float* __restrict__ ws) {
  const int b = blockIdx.y;
  const int h = blockIdx.x * 256 + threadIdx.x;
  __shared__ float s_comb[H_SZ];
  __shared__ float s_h0[H_SZ];
  for (int j = threadIdx.x; j < H_SZ; j += 256) {
    s_comb[j] = ws[WS_COMB + b * H_SZ + j];
    s_h0[j]   = hidden[b * H_SZ + j];
  }
  __syncthreads();
  float ir = b_ih[h], iz = b_ih[H_SZ + h], in = b_ih[2 * H_SZ + h];
  float hr = b_hh[h], hz = b_hh[H_SZ + h], hn = b_hh[2 * H_SZ + h];
  const float* wi0 = W_ih + (size_t)h * H_SZ;
  const float* wi1 = W_ih + (size_t)(H_SZ + h) * H_SZ;
  const float* wi2 = W_ih + (size_t)(2 * H_SZ + h) * H_SZ;
  const float* wh0 = W_hh + (size_t)h * H_SZ;
  const float* wh1 = W_hh + (size_t)(H_SZ + h) * H_SZ;
  const float* wh2 = W_hh + (size_t)(2 * H_SZ + h) * H_SZ;
  #pragma unroll 4
  for (int j = 0; j < H_SZ; ++j) {
    float c = s_comb[j], p = s_h0[j];
    ir = fmaf(c, wi0[j], ir); iz = fmaf(c, wi1[j], iz); in = fmaf(c, wi2[j], in);
    hr = fmaf(p, wh0[j], hr); hz = fmaf(p, wh1[j], hz); hn = fmaf(p, wh2[j], hn);
  }
  float r = sigmoidf_(ir + hr);
  float z = sigmoidf_(iz + hz);
  float n = tanhf(in + r * hn);
  ws[WS_HNEW + b * H_SZ + h] = (1.0f - z) * n + z * s_h0[h];
}

// ---------------- K5: p_gen + hidden passthrough --------------------------
__global__ __launch_bounds__(256) void k5_pgen(
    const float* __restrict__ gen_W, const float* __restrict__ gen_b,
    const float* __restrict__ hidden, float* __restrict__ ws, float* __restrict__ out) {
  const int b = blockIdx.x, tid = threadIdx.x;
  __shared__ float red[256];
  float acc = 0.f;
  for (int j = tid; j < H_SZ; j += 256) {
    acc = fmaf(ws[WS_COMB + b * H_SZ + j], gen_W[j], acc);
    acc = fmaf(hidden[b * H_SZ + j], gen_W[H_SZ + j], acc);
    out[OUT_H + b * H_SZ + j] = hidden[b * H_SZ + j];  // second tuple output
  }
  red[tid] = acc; __syncthreads();
  for (int s = 128; s > 0; s >>= 1) { if (tid < s) red[tid] += red[tid + s]; __syncthreads(); }
  if (tid == 0) ws[WS_PGEN + b] = sigmoidf_(red[0] + gen_b[0]);
}

// ---------------- K6: vocab GEMM (fp32 WMMA 16x16x4) ----------------------
// D(v,b) = sum_k out_W[v,k] * h_new[b,k]; one wave = 16 v-rows x all 64 b.
__global__ __launch_bounds__(256) void k6_vocab_gemm(
    const float* __restrict__ out_W, const float* __restrict__ out_b,
    const float* __restrict__ ws, float* __restrict__ out) {
  __shared__ float tile[8][B_SZ][17];          // [wave][b][v_local], padded
  const int lane = threadIdx.x & 31;
  const int wv   = threadIdx.x >> 5;
  const int hi   = lane >> 4;                  // lane half
  const int n    = lane & 15;                  // A: row M; B: col N
  const int vtile = blockIdx.x * 8 + wv;
  if (vtile >= V_SZ / 16) return;              // wave-uniform, EXEC stays all-1s
  const int v0 = vtile * 16;

  const float* h_new = ws + WS_HNEW;
  const float* aptr  = out_W + (size_t)(v0 + n) * H_SZ + 2 * hi;
  const float* bptr0 = h_new + (size_t)( 0 + n) * H_SZ + 2 * hi;
  const float* bptr1 = h_new + (size_t)(16 + n) * H_SZ + 2 * hi;
  const float* bptr2 = h_new + (size_t)(32 + n) * H_SZ + 2 * hi;
  const float* bptr3 = h_new + (size_t)(48 + n) * H_SZ + 2 * hi;

  v8f c0 = {}, c1 = {}, c2 = {}, c3 = {};
  for (int k = 0; k < H_SZ; k += 4) {
    if ((k & 63) == 0) __builtin_prefetch(aptr + k + 64, 0, 1);   // global_prefetch_b8
    v2f a  = *(const v2f*)(aptr  + k);
    v2f b0 = *(const v2f*)(bptr0 + k);
    v2f b1 = *(const v2f*)(bptr1 + k);
    v2f b2 = *(const v2f*)(bptr2 + k);
    v2f b3 = *(const v2f*)(bptr3 + k);
    c0 = __builtin_amdgcn_wmma_f32_16x16x4_f32(false, a, false, b0, (short)0, c0, false, false);
    c1 = __builtin_amdgcn_wmma_f32_16x16x4_f32(false, a, false, b1, (short)0, c1, false, false);
    c2 = __builtin_amdgcn_wmma_f32_16x16x4_f32(false, a, false, b2, (short)0, c2, false, false);
    c3 = __builtin_amdgcn_wmma_f32_16x16x4_f32(false, a, false, b3, (short)0, c3, false, false);
  }

  // C/D layout: VGPR g holds (M = g + 8*hi, N = lane&15). Stage + bias in LDS.
  #pragma unroll
  for (int g = 0; g < 8; ++g) {
    const int vl = g + 8 * hi;
    const float bias = out_b[v0 + vl];
    tile[wv][ 0 + n][vl] = c0[g] + bias;
    tile[wv][16 + n][vl] = c1[g] + bias;
    tile[wv][32 + n][vl] = c2[g] + bias;
    tile[wv][48 + n][vl] = c3[g] + bias;
  }
  __builtin_amdgcn_wave_barrier();   // intra-wave LDS ordering (DScnt waits inserted)

  // store b-major: each lane writes 2 batch rows x 16 contiguous v (4x float4)
  #pragma unroll
  for (int r = 0; r < 2; ++r) {
    const int bl = lane * 2 + r;
    float* dst = out + OUT_P + (size_t)bl * ROW_P + v0;
    #pragma unroll
    for (int q = 0; q < 16; q += 4) {
      float4 val = make_float4(tile[wv][bl][q], tile[wv][bl][q + 1],
                               tile[wv][bl][q + 2], tile[wv][bl][q + 3]);
      *(float4*)(dst + q) = val;
    }
  }
}

// ---------------- K7: per-row logsumexp over V ----------------------------
__global__ __launch_bounds__(256) void k7_lse(float* __restrict__ out, float* __restrict__ ws) {
  const int b = blockIdx.x, tid = threadIdx.x;
  const float* row = out + OUT_P + (size_t)b * ROW_P;
  __shared__ float red[256];
  float m = -INFINITY;
  for (int v = tid; v < V_SZ; v += 256) m = fmaxf(m, row[v]);
  red[tid] = m; __syncthreads();
  for (int s = 128; s > 0; s >>= 1) { if (tid < s) red[tid] = fmaxf(red[tid], red[tid + s]); __syncthreads(); }
  const float mx = red[0]; __syncthreads();
  float sm = 0.f;
  for (int v = tid; v < V_SZ; v += 256) sm += __expf(row[v] - mx);
  red[tid] = sm; __syncthreads();
  for (int s = 128; s > 0; s >>= 1) { if (tid < s) red[tid] += red[tid + s]; __syncthreads(); }
  if (tid == 0) ws[WS_LSE + b] = mx + __logf(red[0]);
}

// ---------------- K8: p_final = log_softmax * p_gen; zero OOV pad ---------
__global__ __launch_bounds__(256) void k8_finalize(float* __restrict__ out, const float* __restrict__ ws) {
  const int idx = blockIdx.x * 256 + threadIdx.x;       // [0, B*ROW_P)
  const int b = idx / ROW_P;
  const int v = idx - b * ROW_P;
  float* p = out + OUT_P + idx;
  if (v < V_SZ) {
    *p = (*p - ws[WS_LSE + b]) * ws[WS_PGEN + b];
  } else {
    *p = 0.0f;
  }
}

// ---------------- K9: pointer scatter (deterministic, 1 thread / row) -----
__global__ __launch_bounds__(32) void k9_scatter(
    const int* __restrict__ full_input, const float* __restrict__ ws, float* __restrict__ out) {
  const int b = blockIdx.x * 32 + threadIdx.x;
  if (b >= B_SZ) return;
  const float g = 1.0f - ws[WS_PGEN + b];
  float* row = out + OUT_P + (size_t)b * ROW_P;
  const int* fi = full_input + b * L_SZ;
  const float* aw = ws + WS_ATTN + b * L_SZ;
  for (int l = 0; l < L_SZ; ++l) row[fi[l]] += g * aw[l];
}

extern "C" void kernel_launch(void* const* d_in, const int* in_sizes, int n_in,
                              void* d_out, int out_size, void* d_ws, size_t ws_size,
                              hipStream_t stream) {
  (void)in_sizes; (void)n_in; (void)out_size; (void)ws_size;
  const int*   input_ids = (const int*)  d_in[0];
  const float* hidden    = (const float*)d_in[1];
  const float* enc       = (const float*)d_in[2];
  const int*   full_in   = (const int*)  d_in[3];
  const float* emb       = (const float*)d_in[4];
  const float* attn_W    = (const float*)d_in[5];
  const float* attn_b    = (const float*)d_in[6];
  const float* comb_W    = (const float*)d_in[7];
  const float* comb_b    = (const float*)d_in[8];
  const float* W_ih      = (const float*)d_in[9];
  const float* W_hh      = (const float*)d_in[10];
  const float* b_ih      = (const float*)d_in[11];
  const float* b_hh      = (const float*)d_in[12];
  const float* out_W     = (const float*)d_in[13];
  const float* out_b     = (const float*)d_in[14];
  const float* gen_W     = (const float*)d_in[15];
  const float* gen_b     = (const float*)d_in[16];
  float* out = (float*)d_out;
  float* ws  = (float*)d_ws;

  k1_front_attn<<<B_SZ, 256, 0, stream>>>(input_ids, hidden, emb, attn_W, attn_b, ws, out);
  k2_attn_apply<<<dim3(H_SZ / 256, B_SZ), 256, 0, stream>>>(enc, ws, ws);
  k3_combine   <<<dim3(H_SZ / 256, B_SZ), 256, 0, stream>>>(comb_W, comb_b, ws);
  k4_gru       <<<dim3(H_SZ / 256, B_SZ), 256, 0, stream>>>(W_ih, W_hh, b_ih, b_hh, hidden, ws);
  k5_pgen      <<<B_SZ, 256, 0, stream>>>(gen_W, gen_b, hidden, ws, out);
  k6_vocab_gemm<<<(V_SZ / 16 + 7) / 8, 256, 0, stream>>>(out_W, out_b, ws, out);
  k7_lse       <<<B_SZ, 256, 0, stream>>>(out, ws);
  k8_finalize  <<<(B_SZ * ROW_P) / 256, 256, 0, stream>>>(out, ws);
  k9_scatter   <<<(B_SZ + 31) / 32, 32, 0, stream>>>(full_in, ws, out);
}